// PointerLSTM_44272522887215
// MI455X (gfx1250) — compile-verified
//
#include <hip/hip_runtime.h>
#include <stdint.h>

#define Bb 128
#define Tt 128
#define Dd 256
#define Hh 256
#define Z4 1024  // 4*H

typedef __attribute__((ext_vector_type(2))) float v2f;
typedef __attribute__((ext_vector_type(8))) float v8f;

__device__ __forceinline__ float sigmoidf_(float x) {
  return 1.0f / (1.0f + __expf(-x));
}

// bf16 round-to-nearest-even (finite inputs)
__device__ __forceinline__ uint32_t bf16_rne(float f) {
  uint32_t u = __float_as_uint(f);
  return (u + 0x7fffu + ((u >> 16) & 1u)) >> 16;
}

// ---------------------------------------------------------------------------
// C[M,N] = A[M,K](lda) @ B[K,N](ldb) (+ bias[N]), fp32 via V_WMMA_F32_16X16X4_F32.
// One wave computes one 16x16 tile of C; K-loop in steps of 4.
// ---------------------------------------------------------------------------
__global__ __launch_bounds__(256) void wmma_gemm_f32(
    const float* __restrict__ A, int lda,
    const float* __restrict__ Bm, int ldb,
    const float* __restrict__ bias,
    float* __restrict__ C, int ldc,
    int M, int N, int K)
{
  const int wavesPerBlock = blockDim.x >> 5;
  int waveId = blockIdx.x * wavesPerBlock + (threadIdx.x >> 5);
  int tilesN = N >> 4;
  int tilesM = M >> 4;
  if (waveId >= tilesM * tilesN) return;   // wave-uniform branch, EXEC stays all-ones
  int tm = waveId / tilesN;
  int tn = waveId - tm * tilesN;
  int lane = threadIdx.x & 31;
  int half = lane >> 4;
  int lm   = lane & 15;
  const float* Arow = A + (size_t)(tm * 16 + lm) * lda;
  int col = tn * 16 + lm;

  v8f acc = {};
  for (int k0 = 0; k0 < K; k0 += 4) {
    int ka = k0 + 2 * half;
    v2f a, b;
    a.x = Arow[ka];
    a.y = Arow[ka + 1];
    b.x = Bm[(size_t)ka * ldb + col];
    b.y = Bm[(size_t)(ka + 1) * ldb + col];
    acc = __builtin_amdgcn_wmma_f32_16x16x4_f32(
        false, a, false, b, (short)0, acc, false, false);
  }

  float bb = bias ? bias[col] : 0.0f;
  int crow = tm * 16 + 8 * half;
  #pragma unroll
  for (int r = 0; r < 8; ++r) {
    C[(size_t)(crow + r) * ldc + col] = acc[r] + bb;
  }
}

// ---------------------------------------------------------------------------
// Pack W(K x N, fp32) into P((K/2) x N, u32): P[kp*N+n] = {bf16(W[2kp+1][n]) << 16 | bf16(W[2kp][n])}
// Halves the per-step weight-stream bytes in the scan (the dominant memory cost).
// ---------------------------------------------------------------------------
__global__ __launch_bounds__(256) void pack_bf16_pairs(
    const float* __restrict__ W, uint32_t* __restrict__ P, int N, int K)
{
  int idx = blockIdx.x * blockDim.x + threadIdx.x;
  int total = (K >> 1) * N;
  if (idx >= total) return;
  int kp = idx / N;
  int n  = idx - kp * N;
  uint32_t b0 = bf16_rne(W[(size_t)(2 * kp) * N + n]);
  uint32_t b1 = bf16_rne(W[(size_t)(2 * kp + 1) * N + n]);
  P[idx] = (b1 << 16) | b0;
}

// ---------------------------------------------------------------------------
// One workgroup per batch element b; runs the whole 128-step scan internally.
// LDS: w2_en[b] (128 KB) staged once via async-to-LDS + small buffers (~140 KB).
// ---------------------------------------------------------------------------
__global__ __launch_bounds__(1024) void pointer_lstm_scan(
    const float*    __restrict__ xz,    // B x 1024
    const float*    __restrict__ w2en,  // B x T x H
    const uint32_t* __restrict__ Rp,    // (H/2) x 4H  (bf16 k-pairs)
    const uint32_t* __restrict__ W1p,   // (H/2) x H   (bf16 k-pairs)
    const float*    __restrict__ V,     // H
    float*          __restrict__ out)   // B x T x T : out[b, step, t]
{
  extern __shared__ float smem[];
  float* w2s    = smem;                 // T*H = 32768
  float* xzs    = w2s + Tt * Hh;        // 1024
  float* zs     = xzs + Z4;             // 1024
  float* hs     = zs + Z4;              // 256
  float* s1s    = hs + Hh;              // 256
  float* Vs     = s1s + Hh;             // 256
  float* scores = Vs + Hh;              // 128

  const int b    = blockIdx.x;
  const int tid  = threadIdx.x;
  const int wave = tid >> 5;
  const int lane = tid & 31;

  // Stage w2_en[b] into LDS with CDNA5 async copies (1024 lanes x 16B x 8 iters = 128 KB).
  const float* w2b = w2en + (size_t)b * Tt * Hh;
  #pragma unroll
  for (int it = 0; it < (Tt * Hh) / (1024 * 4); ++it) {
    int i = tid * 4 + it * 1024 * 4;
    uint32_t lds_addr = (uint32_t)(uintptr_t)(w2s + i);   // low 32 bits == LDS offset
    uint64_t gaddr    = (uint64_t)(uintptr_t)(w2b + i);
    asm volatile("global_load_async_to_lds_b128 %0, %1, off"
                 :: "v"(lds_addr), "v"(gaddr) : "memory");
  }
  xzs[tid] = xz[(size_t)b * Z4 + tid];
  if (tid < Hh) { Vs[tid] = V[tid]; hs[tid] = 0.0f; }
  float c_reg = 0.0f;
  asm volatile("s_wait_asynccnt 0" ::: "memory");
  __syncthreads();

  for (int step = 0; step < Tt; ++step) {
    // z[tid] = xz[tid] + sum_k h[k] * R[k][tid]   (bf16-packed R: 4B -> 2 FMAs)
    float acc = xzs[tid];
    #pragma unroll 8
    for (int kp = 0; kp < Hh / 2; ++kp) {
      uint32_t w = Rp[(size_t)kp * Z4 + tid];
      float f0 = __uint_as_float(w << 16);
      float f1 = __uint_as_float(w & 0xffff0000u);
      acc += hs[2 * kp] * f0;
      acc += hs[2 * kp + 1] * f1;
    }
    zs[tid] = acc;
    __syncthreads();

    // gate math; carry c in a register of thread j (<256), h to LDS
    if (tid < Hh) {
      float ig = sigmoidf_(zs[tid]);
      float fg = sigmoidf_(zs[tid + Hh]);
      float gg = tanhf(zs[tid + 2 * Hh]);
      float og = sigmoidf_(zs[tid + 3 * Hh]);
      c_reg = fg * c_reg + ig * gg;
      hs[tid] = og * tanhf(c_reg);
    }
    __syncthreads();

    // s1 = h_new @ W1  (threads 0..255, bf16-packed W1)
    if (tid < Hh) {
      float a2 = 0.0f;
      #pragma unroll 8
      for (int kp = 0; kp < Hh / 2; ++kp) {
        uint32_t w = W1p[(size_t)kp * Hh + tid];
        float f0 = __uint_as_float(w << 16);
        float f1 = __uint_as_float(w & 0xffff0000u);
        a2 += hs[2 * kp] * f0;
        a2 += hs[2 * kp + 1] * f1;
      }
      s1s[tid] = a2;
    }
    __syncthreads();

    // attention scores: one wave per row t (conflict-free LDS banks: lane+32*i)
    for (int t = wave; t < Tt; t += 32) {
      const float* row = w2s + t * Hh;
      float p = 0.0f;
      #pragma unroll
      for (int i = 0; i < Hh / 32; ++i) {
        int hh = lane + 32 * i;
        p += tanhf(s1s[hh] + row[hh]) * Vs[hh];
      }
      #pragma unroll
      for (int off = 16; off > 0; off >>= 1) p += __shfl_down(p, off, 32);
      if (lane == 0) scores[t] = p;
    }
    __syncthreads();

    // softmax over t (single wave), write probs to out[b, step, :]
    if (wave == 0) {
      float m = -3.4e38f;
      #pragma unroll
      for (int i = 0; i < Tt / 32; ++i) m = fmaxf(m, scores[lane + 32 * i]);
      #pragma unroll
      for (int off = 16; off > 0; off >>= 1) m = fmaxf(m, __shfl_xor(m, off, 32));
      float e[Tt / 32];
      float ssum = 0.0f;
      #pragma unroll
      for (int i = 0; i < Tt / 32; ++i) {
        e[i] = __expf(scores[lane + 32 * i] - m);
        ssum += e[i];
      }
      #pragma unroll
      for (int off = 16; off > 0; off >>= 1) ssum += __shfl_xor(ssum, off, 32);
      float inv = 1.0f / ssum;
      float* op = out + (size_t)b * Tt * Tt + (size_t)step * Tt;
      #pragma unroll
      for (int i = 0; i < Tt / 32; ++i) op[lane + 32 * i] = e[i] * inv;
    }
    __syncthreads();
  }
}

extern "C" void kernel_launch(void* const* d_in, const int* in_sizes, int n_in,
                              void* d_out, int out_size, void* d_ws, size_t ws_size,
                              hipStream_t stream) {
  (void)in_sizes; (void)n_in; (void)out_size; (void)ws_size;
  const float* x    = (const float*)d_in[0];  // (B, T, D)
  const float* kern = (const float*)d_in[1];  // (D, 4H)
  const float* R    = (const float*)d_in[2];  // (H, 4H)
  const float* bias = (const float*)d_in[3];  // (4H)
  const float* W1   = (const float*)d_in[4];  // (H, H)
  const float* W2   = (const float*)d_in[5];  // (D, H)
  const float* V    = (const float*)d_in[6];  // (H, 1)
  float* out = (float*)d_out;                 // (B, T, T)

  float*    xz   = (float*)d_ws;                      // B * 4H
  float*    w2en = xz + (size_t)Bb * Z4;              // B * T * H
  uint32_t* Rp   = (uint32_t*)(w2en + (size_t)Bb * Tt * Hh);  // (H/2) * 4H
  uint32_t* W1p  = Rp + (size_t)(Hh / 2) * Z4;                // (H/2) * H

  // Pack recurrent weights to bf16 k-pairs (halves scan weight traffic).
  {
    int totalR = (Hh / 2) * Z4;     // 131072
    pack_bf16_pairs<<<(totalR + 255) / 256, 256, 0, stream>>>(R, Rp, Z4, Hh);
    int totalW = (Hh / 2) * Hh;     // 32768
    pack_bf16_pairs<<<(totalW + 255) / 256, 256, 0, stream>>>(W1, W1p, Hh, Hh);
  }
  // w2_en = x.reshape(B*T, D) @ W2
  {
    int M = Bb * Tt, N = Hh, K = Dd;
    int tiles = (M / 16) * (N / 16);
    wmma_gemm_f32<<<(tiles + 7) / 8, 256, 0, stream>>>(x, Dd, W2, Hh, nullptr, w2en, Hh, M, N, K);
  }
  // xz = x[:, -1, :] @ kernel + bias  (A = strided view, lda = T*D)
  {
    int M = Bb, N = Z4, K = Dd;
    int tiles = (M / 16) * (N / 16);
    wmma_gemm_f32<<<(tiles + 7) / 8, 256, 0, stream>>>(x + (size_t)(Tt - 1) * Dd, Tt * Dd,
                                                       kern, Z4, bias, xz, Z4, M, N, K);
  }
  // Whole 128-step scan in one kernel: one workgroup per batch element.
  size_t smem = (size_t)(Tt * Hh + Z4 + Z4 + Hh + Hh + Hh + Tt) * sizeof(float);
  pointer_lstm_scan<<<Bb, 1024, smem, stream>>>(xz, w2en, Rp, W1p, V, out);
}